// Model_42219528519938
// MI455X (gfx1250) — compile-verified
//
#include <hip/hip_runtime.h>
#include <math.h>

// ---------------------------------------------------------------------------
// Problem constants (match reference)
// ---------------------------------------------------------------------------
#define D_MAIN    256
#define D_INT     512
#define N_FEAT    128
#define N_CLASSES 10
#define CTX       96
#define NPART     16               // candidate partitions for top-k occupancy
#define PAD       4
#define LDA       (D_MAIN + PAD)   // 260 floats per LDS row (16B-aligned stride)
#define LDF       (N_FEAT + PAD)   // 132
#define LDH       68               // 64 + 4

typedef float v2f __attribute__((ext_vector_type(2)));
typedef float v8f __attribute__((ext_vector_type(8)));

// ---------------------------------------------------------------------------
// Tensor Data Mover support (guarded: fallback = plain vector loads)
// ---------------------------------------------------------------------------
#if defined(__has_builtin)
#if __has_builtin(__builtin_amdgcn_tensor_load_to_lds) && \
    __has_builtin(__builtin_amdgcn_s_wait_tensorcnt)
#define HAVE_TDM 1
#endif
#endif
#ifndef HAVE_TDM
#define HAVE_TDM 0
#endif

#if HAVE_TDM
typedef unsigned int v4u __attribute__((ext_vector_type(4)));
typedef int          v4i __attribute__((ext_vector_type(4)));
typedef int          v8i __attribute__((ext_vector_type(8)));

// Issue one TDM load of a 64 x 256 f32 tile (rows beyond rows_avail read as 0),
// into LDS with a 260-float row stride (pad 4 DWORDs after every 256 DWORDs).
__device__ __forceinline__ void tdm_load_tile(void* lds_dst,
                                              const float* gsrc,
                                              int rows_avail)
{
  const unsigned lds_off = (unsigned)(size_t)lds_dst;  // LDS addr = low 32 bits
  const unsigned long long ga = (unsigned long long)(size_t)gsrc;
  const unsigned rem = (unsigned)rows_avail;

  v4u g0;
  g0[0] = 1u;                                   // count = 1 (valid descriptor)
  g0[1] = lds_off;                              // lds_addr
  g0[2] = (unsigned)ga;                         // global_addr[31:0]
  g0[3] = (unsigned)((ga >> 32) & 0x01FFFFFFull) | (2u << 30);  // addr hi | type=2

  v8i g1;
  g1[0] = (int)((2u << 16) |                    // data_size = 4B
                (1u << 20) |                    // pad_enable
                (7u << 22) |                    // pad_interval: 256 DWORDs
                (3u << 25));                    // pad_amount: 4 DWORDs
  g1[1] = (int)(256u << 16);                    // tensor_dim0 = 256 (lo16)
  g1[2] = (int)((rem & 0xFFFFu) << 16);         // td0 hi16=0 | tensor_dim1 lo16
  g1[3] = (int)((rem >> 16) | (256u << 16));    // td1 hi16   | tile_dim0 = 256
  g1[4] = 64;                                   // tile_dim1 = 64, tile_dim2 = 0
  g1[5] = 256;                                  // tensor_dim0_stride = 256
  g1[6] = 0;
  g1[7] = 0;

  v4i gz; gz[0] = 0; gz[1] = 0; gz[2] = 0; gz[3] = 0;
#if __clang_major__ >= 23
  v8i gz8;
#pragma unroll
  for (int i = 0; i < 8; ++i) gz8[i] = 0;
  __builtin_amdgcn_tensor_load_to_lds(g0, g1, gz, gz, gz8, 0);
#else
  __builtin_amdgcn_tensor_load_to_lds(g0, g1, gz, gz, 0);
#endif
}
#endif  // HAVE_TDM

// ---------------------------------------------------------------------------
// WMMA helpers: D(16x16,f32) += A(16xK) * B(Kx16), K multiple of 4.
// V_WMMA_F32_16X16X4_F32 fragment layout (wave32):
//   A: lane l (0-15): v0=A[l][k0+0], v1=A[l][k0+1]; lane l+16: K+2/K+3
//   B: lane l (0-15): v0=B[k0+0][l], v1=B[k0+1][l]; lane l+16: K+2/K+3
//   C/D: vgpr r, lanes 0-15 -> (M=r, N=l); lanes 16-31 -> (M=r+8, N=l)
// ---------------------------------------------------------------------------
__device__ __forceinline__ v8f wmma_AB(const float* __restrict__ As, int lda,
                                       const float* __restrict__ Bg, int ldb,
                                       int K, v8f acc, int lane)
{
  const int half = lane >> 4;
  const int l    = lane & 15;
  const float* arow = As + (size_t)l * lda;
#pragma unroll 4
  for (int k0 = 0; k0 < K; k0 += 4) {
    const int ka = k0 + 2 * half;
    v2f a, b;
    a.x = arow[ka];
    a.y = arow[ka + 1];
    b.x = Bg[(size_t)ka * ldb + l];
    b.y = Bg[(size_t)(ka + 1) * ldb + l];
    acc = __builtin_amdgcn_wmma_f32_16x16x4_f32(false, a, false, b,
                                                (short)0, acc, false, false);
  }
  return acc;
}

// B supplied transposed: element B[k][n] = Bt[n*ldbt + k]  (for k @ ki^T)
__device__ __forceinline__ v8f wmma_A_Bt(const float* __restrict__ As, int lda,
                                         const float* __restrict__ Bt, int ldbt,
                                         int K, v8f acc, int lane)
{
  const int half = lane >> 4;
  const int l    = lane & 15;
  const float* arow = As + (size_t)l * lda;
  const float* brow = Bt + (size_t)l * ldbt;
#pragma unroll 4
  for (int k0 = 0; k0 < K; k0 += 4) {
    const int ka = k0 + 2 * half;
    v2f a, b;
    a.x = arow[ka];
    a.y = arow[ka + 1];
    b.x = brow[ka];
    b.y = brow[ka + 1];
    acc = __builtin_amdgcn_wmma_f32_16x16x4_f32(false, a, false, b,
                                                (short)0, acc, false, false);
  }
  return acc;
}

// ---------------------------------------------------------------------------
// Kernel 1: fused two-stage row encoder.
//   x = in(16xN_FEAT) @ W1(128x256) + b1          (kept in LDS, optional store)
//   k = x @ W2(256x256) + b2                      (stored to k_out)
// block = 128 threads (4 waves), block handles 16 rows.
// ---------------------------------------------------------------------------
__global__ void __launch_bounds__(128)
encode_kernel(const float* __restrict__ in,
              const float* __restrict__ W1, const float* __restrict__ b1,
              const float* __restrict__ W2, const float* __restrict__ b2,
              float* __restrict__ x_out, float* __restrict__ k_out, int nrows)
{
  __shared__ float sIn[16][LDF];
  __shared__ float sX[16][LDA];
  const int tid  = threadIdx.x;
  const int lane = tid & 31;
  const int wid  = tid >> 5;
  const int half = lane >> 4;
  const int l    = lane & 15;
  const int row0 = blockIdx.x * 16;

  const float4* in4 = (const float4*)(in + (size_t)row0 * N_FEAT);
  for (int i = tid; i < 16 * (N_FEAT / 4); i += 128) {
    const int r = i >> 5, c4 = i & 31;
    float4 v = in4[r * 32 + c4];
    *(float4*)&sIn[r][c4 * 4] = v;
  }
  __syncthreads();

#pragma unroll
  for (int ct = 0; ct < 4; ++ct) {
    const int col0 = wid * 64 + ct * 16;
    const float bb = b1[col0 + l];
    v8f acc;
#pragma unroll
    for (int r = 0; r < 8; ++r) acc[r] = bb;
    acc = wmma_AB(&sIn[0][0], LDF, W1 + col0, D_MAIN, N_FEAT, acc, lane);
#pragma unroll
    for (int r = 0; r < 8; ++r) sX[r + 8 * half][col0 + l] = acc[r];
  }
  __syncthreads();

#pragma unroll
  for (int ct = 0; ct < 4; ++ct) {
    const int col0 = wid * 64 + ct * 16;
    const float bb = b2[col0 + l];
    v8f acc;
#pragma unroll
    for (int r = 0; r < 8; ++r) acc[r] = bb;
    acc = wmma_AB(&sX[0][0], LDA, W2 + col0, D_MAIN, D_MAIN, acc, lane);
#pragma unroll
    for (int r = 0; r < 8; ++r)
      k_out[(size_t)(row0 + r + 8 * half) * D_MAIN + col0 + l] = acc[r];
  }

  if (x_out) {  // uniform branch (kernel argument)
    __syncthreads();
    for (int i = tid; i < 16 * (D_MAIN / 4); i += 128) {
      const int r = i >> 6, c4 = i & 63;
      float4 v = *(const float4*)&sX[r][c4 * 4];
      ((float4*)(x_out + (size_t)(row0 + r) * D_MAIN))[c4] = v;
    }
  }
}

// ---------------------------------------------------------------------------
// Kernel 2: per-row squared L2 norm of a [n x 256] matrix.
// ---------------------------------------------------------------------------
__global__ void row_sumsq_kernel(const float* __restrict__ M,
                                 float* __restrict__ out, int nrows)
{
  const int row = blockIdx.x * blockDim.x + threadIdx.x;
  if (row >= nrows) return;
  const float4* r4 = (const float4*)(M + (size_t)row * D_MAIN);
  float s = 0.f;
#pragma unroll 8
  for (int i = 0; i < D_MAIN / 4; ++i) {
    float4 v = r4[i];
    s += v.x * v.x + v.y * v.y + v.z * v.z + v.w * v.w;
  }
  out[row] = s;
}

// ---------------------------------------------------------------------------
// Kernel 3a: partitioned fused distance GEMM + streaming top-96.
// grid = (nqueries/16, NPART); block = 128 threads (4 waves) -> 1024 WGs.
// Each block scans its candidate slice in 64-row chunks staged into LDS by
// the Tensor Data Mover (double-buffered; DMA of chunk i+1 overlaps WMMA of
// chunk i), computes ||ki||^2 - 2 k.ki (query-norm constant dropped), and a
// serial selector thread per query keeps the slice-local top-96.
// ---------------------------------------------------------------------------
__global__ void __launch_bounds__(128)
topk_part_kernel(const float* __restrict__ kq, const float* __restrict__ ki,
                 const float* __restrict__ cn, float* __restrict__ pdist,
                 int* __restrict__ pidx, int ncand, int nparts)
{
  extern __shared__ char smem[];
  float (*sK)[LDA]  = (float (*)[LDA])smem;                            // 16x260
  float (*sKi)[LDA] = (float (*)[LDA])(smem + 16 * LDA * 4);           // 128x260
  float (*sD)[LDH]  = (float (*)[LDH])(smem + (16 + 128) * LDA * 4);   // 16x68
  float* sList = (float*)(smem + (16 + 128) * LDA * 4 + 16 * LDH * 4); // 16*96
  int*   sIdx  = (int*)(sList + 16 * CTX);                             // 16*96

  const int tid  = threadIdx.x;
  const int lane = tid & 31;
  const int wid  = tid >> 5;
  const int half = lane >> 4;
  const int l    = lane & 15;
  const int q0   = blockIdx.x * 16;

  const int span  = (ncand + nparts - 1) / nparts;
  const int pbase = blockIdx.y * span;
  const int pend  = min(pbase + span, ncand);
  const int nch   = (pend - pbase + 63) / 64;

  const float4* kq4 = (const float4*)(kq + (size_t)q0 * D_MAIN);
  for (int i = tid; i < 16 * (D_MAIN / 4); i += 128) {
    const int r = i >> 6, c4 = i & 63;
    *(float4*)&sK[r][c4 * 4] = kq4[r * 64 + c4];
  }

  int   cnt  = 0, carg = 0;
  float cmax = -1e30f;

#if HAVE_TDM
  if (wid == 0)  // one wave issues DMA; TENSORcnt is per-wave
    tdm_load_tile(&sKi[0][0], ki + (size_t)pbase * D_MAIN, pend - pbase);
#endif

  for (int ch = 0; ch < nch; ++ch) {
    const int base = pbase + ch * 64;
#if HAVE_TDM
    const int cur = ch & 1;
    if (wid == 0) __builtin_amdgcn_s_wait_tensorcnt(0);
    __syncthreads();  // chunk `ch` visible in sKi[cur*64 ..]
    if (wid == 0 && ch + 1 < nch)
      tdm_load_tile(&sKi[(cur ^ 1) * 64][0],
                    ki + (size_t)(base + 64) * D_MAIN, pend - (base + 64));
#else
    const int cur = 0;
    for (int i = tid; i < 64 * (D_MAIN / 4); i += 128) {
      const int rr = i >> 6, c4 = i & 63;
      const int g = base + rr;
      float4 v = make_float4(0.f, 0.f, 0.f, 0.f);
      if (g < pend) v = ((const float4*)(ki + (size_t)g * D_MAIN))[c4];
      *(float4*)&sKi[rr][c4 * 4] = v;
      if (g + 64 < pend)  // warm caches for next chunk -> global_prefetch_b8
        __builtin_prefetch(ki + (size_t)(g + 64) * D_MAIN + c4 * 4, 0, 1);
    }
    __syncthreads();
#endif

    // one 16x16 dot-product tile per wave (K = 256)
    const int n0 = wid * 16;
    v8f acc;
#pragma unroll
    for (int r = 0; r < 8; ++r) acc[r] = 0.f;
    acc = wmma_A_Bt(&sK[0][0], LDA, &sKi[cur * 64 + n0][0], LDA,
                    D_MAIN, acc, lane);
    {
      const int cg = base + n0 + l;
      const float cnv = (cg < pend) ? cn[cg] : 0.f;
#pragma unroll
      for (int r = 0; r < 8; ++r) {
        const float dv = (cg < pend) ? (cnv - 2.0f * acc[r]) : 1e30f;
        sD[r + 8 * half][n0 + l] = dv;
      }
    }
    __syncthreads();

    if (tid < 16) {  // serial selector per query
      float* myL = sList + tid * CTX;
      int*   myI = sIdx + tid * CTX;
      for (int j = 0; j < 64; ++j) {
        const int cg = base + j;
        if (cg >= pend) break;
        const float dj = sD[tid][j];
        if (cnt < CTX) {
          myL[cnt] = dj; myI[cnt] = cg;
          if (cnt == 0 || dj > cmax) { cmax = dj; carg = cnt; }
          ++cnt;
        } else if (dj < cmax) {
          myL[carg] = dj; myI[carg] = cg;
          cmax = -1e30f;
          for (int t = 0; t < CTX; ++t)
            if (myL[t] > cmax) { cmax = myL[t]; carg = t; }
        }
      }
    }
    __syncthreads();
  }

  if (tid < 16) {
    float* myL = sList + tid * CTX;
    int*   myI = sIdx + tid * CTX;
    for (int t = cnt; t < CTX; ++t) { myL[t] = 1e30f; myI[t] = 0; }  // padding
    const size_t o = ((size_t)(q0 + tid) * nparts + blockIdx.y) * CTX;
    for (int t = 0; t < CTX; ++t) { pdist[o + t] = myL[t]; pidx[o + t] = myI[t]; }
  }
}

// ---------------------------------------------------------------------------
// Kernel 3b: merge the NPART per-partition top-96 lists into the global
// top-96 (exact: each global winner is in its partition's top-96).
// ---------------------------------------------------------------------------
__global__ void __launch_bounds__(64)
topk_merge_kernel(const float* __restrict__ pdist, const int* __restrict__ pidx,
                  int* __restrict__ topi, int nparts)
{
  __shared__ float sL[CTX];
  __shared__ int   sI[CTX];
  const int b = blockIdx.x;
  if (threadIdx.x == 0) {
    int cnt = 0, carg = 0;
    float cmax = -1e30f;
    const float* pd = pdist + (size_t)b * nparts * CTX;
    const int*   pi = pidx + (size_t)b * nparts * CTX;
    for (int t = 0; t < nparts * CTX; ++t) {
      const float dj = pd[t];
      if (dj >= 1e30f) continue;  // partition padding
      const int cg = pi[t];
      if (cnt < CTX) {
        sL[cnt] = dj; sI[cnt] = cg;
        if (cnt == 0 || dj > cmax) { cmax = dj; carg = cnt; }
        ++cnt;
      } else if (dj < cmax) {
        sL[carg] = dj; sI[carg] = cg;
        cmax = -1e30f;
        for (int u = 0; u < CTX; ++u)
          if (sL[u] > cmax) { cmax = sL[u]; carg = u; }
      }
    }
    for (int t = 0; t < CTX; ++t) topi[(size_t)b * CTX + t] = sI[t];
  }
}

// ---------------------------------------------------------------------------
// Kernel 4: per-query context aggregation.
//   diff = k_b - ki[idx]; w = softmax(||diff||^2)
//   V    = relu(diff@T_W1 + b1)@T_W2  (chunked over D_INT, V in register frags)
//   x2_b = x_b + sum_c w_c (V_c + Y_emb[y_c])
// ---------------------------------------------------------------------------
__global__ void __launch_bounds__(256)
aggregate_kernel(const float* __restrict__ xq, const float* __restrict__ kq,
                 const float* __restrict__ ki, const int* __restrict__ topi,
                 const int* __restrict__ cand_y, const float* __restrict__ Yemb,
                 const float* __restrict__ TW1, const float* __restrict__ Tb1,
                 const float* __restrict__ TW2, float* __restrict__ x2)
{
  extern __shared__ char smem[];
  float (*sDiff)[LDA] = (float (*)[LDA])smem;                    // 96 x 260
  float (*sH)[LDH]    = (float (*)[LDH])(smem + 96 * LDA * 4);   // 96 x 68
  float* sKb = (float*)(smem + 96 * LDA * 4 + 96 * LDH * 4);     // 256
  float* sS  = sKb + D_MAIN;                                     // 96
  float* sW  = sS + CTX;                                         // 96
  int*   sI  = (int*)(sW + CTX);                                 // 96
  int*   sY  = sI + CTX;                                         // 96
  float* sRed = (float*)(sY + CTX);                              // 2

  const int tid  = threadIdx.x;
  const int lane = tid & 31;
  const int wid  = tid >> 5;
  const int half = lane >> 4;
  const int l    = lane & 15;
  const int b    = blockIdx.x;

  if (tid < D_MAIN) sKb[tid] = kq[(size_t)b * D_MAIN + tid];
  if (tid < CTX) {
    const int id = topi[(size_t)b * CTX + tid];
    sI[tid] = id;
    sY[tid] = cand_y[id];
  }
  __syncthreads();

  for (int i = tid; i < CTX * D_MAIN; i += 256) {
    const int c = i >> 8, d = i & 255;
    sDiff[c][d] = sKb[d] - ki[(size_t)sI[c] * D_MAIN + d];
  }
  __syncthreads();

  if (tid < CTX) {
    float s = 0.f;
    for (int d = 0; d < D_MAIN; ++d) { const float v = sDiff[tid][d]; s += v * v; }
    sS[tid] = s;
  }
  __syncthreads();
  if (tid == 0) {
    float m = sS[0];
    for (int c = 1; c < CTX; ++c) m = fmaxf(m, sS[c]);
    sRed[0] = m;
  }
  __syncthreads();
  if (tid < CTX) sW[tid] = __expf(sS[tid] - sRed[0]);
  __syncthreads();
  if (tid == 0) {
    float s = 0.f;
    for (int c = 0; c < CTX; ++c) s += sW[c];
    sRed[1] = s;
  }
  __syncthreads();
  if (tid < CTX) sW[tid] = sW[tid] / sRed[1];

  v8f vacc[12];
#pragma unroll
  for (int i = 0; i < 12; ++i)
#pragma unroll
    for (int r = 0; r < 8; ++r) vacc[i][r] = 0.f;

  for (int c0 = 0; c0 < D_INT; c0 += 64) {
    for (int t = wid; t < 24; t += 8) {
      const int m0 = (t >> 2) << 4;
      const int n0 = (t & 3) << 4;
      const float bb = Tb1[c0 + n0 + l];
      v8f acc;
#pragma unroll
      for (int r = 0; r < 8; ++r) acc[r] = bb;
      acc = wmma_AB(&sDiff[m0][0], LDA, TW1 + c0 + n0, D_INT, D_MAIN, acc, lane);
#pragma unroll
      for (int r = 0; r < 8; ++r)
        sH[m0 + r + 8 * half][n0 + l] = fmaxf(acc[r], 0.f);
    }
    __syncthreads();
#pragma unroll
    for (int i = 0; i < 12; ++i) {
      const int t  = wid + (i << 3);
      const int m0 = (t >> 4) << 4;
      const int n0 = (t & 15) << 4;
      vacc[i] = wmma_AB(&sH[m0][0], LDH, TW2 + (size_t)c0 * D_MAIN + n0,
                        D_MAIN, 64, vacc[i], lane);
    }
    __syncthreads();
  }

#pragma unroll
  for (int i = 0; i < 12; ++i) {
    const int t  = wid + (i << 3);
    const int m0 = (t >> 4) << 4;
    const int n0 = (t & 15) << 4;
#pragma unroll
    for (int r = 0; r < 8; ++r)
      sDiff[m0 + r + 8 * half][n0 + l] = vacc[i][r];
  }
  __syncthreads();

  if (tid < D_MAIN) {
    const int d = tid;
    float acc = xq[(size_t)b * D_MAIN + d];
    for (int c = 0; c < CTX; ++c)
      acc += sW[c] * (sDiff[c][d] + Yemb[(size_t)sY[c] * D_MAIN + d]);
    x2[(size_t)b * D_MAIN + d] = acc;
  }
}

// ---------------------------------------------------------------------------
// Kernel 5: predictor block + head (tiny: plain VALU). One block per query.
// ---------------------------------------------------------------------------
__global__ void __launch_bounds__(256)
head_kernel(const float* __restrict__ x2,
            const float* __restrict__ ln1s, const float* __restrict__ ln1b,
            const float* __restrict__ W1, const float* __restrict__ b1,
            const float* __restrict__ W2, const float* __restrict__ b2,
            const float* __restrict__ ln2s, const float* __restrict__ ln2b,
            const float* __restrict__ PW, const float* __restrict__ Pb,
            float* __restrict__ out)
{
  __shared__ float sx[D_MAIN], sln[D_MAIN], sh[D_INT], sx3[D_MAIN], sr[D_MAIN];
  __shared__ float sms[2];
  const int tid = threadIdx.x;
  const int b   = blockIdx.x;

  sx[tid] = x2[(size_t)b * D_MAIN + tid];
  __syncthreads();
  if (tid == 0) {
    float m = 0.f;
    for (int i = 0; i < D_MAIN; ++i) m += sx[i];
    m /= D_MAIN;
    float v = 0.f;
    for (int i = 0; i < D_MAIN; ++i) { const float d = sx[i] - m; v += d * d; }
    v /= D_MAIN;
    sms[0] = m; sms[1] = rsqrtf(v + 1e-5f);
  }
  __syncthreads();
  sln[tid] = (sx[tid] - sms[0]) * sms[1] * ln1s[tid] + ln1b[tid];
  __syncthreads();
  for (int o = tid; o < D_INT; o += 256) {
    float a = b1[o];
    for (int k = 0; k < D_MAIN; ++k) a += sln[k] * W1[(size_t)k * D_INT + o];
    sh[o] = fmaxf(a, 0.f);
  }
  __syncthreads();
  {
    float a = b2[tid];
    for (int k = 0; k < D_INT; ++k) a += sh[k] * W2[(size_t)k * D_MAIN + tid];
    sx3[tid] = sx[tid] + a;
  }
  __syncthreads();
  if (tid == 0) {
    float m = 0.f;
    for (int i = 0; i < D_MAIN; ++i) m += sx3[i];
    m /= D_MAIN;
    float v = 0.f;
    for (int i = 0; i < D_MAIN; ++i) { const float d = sx3[i] - m; v += d * d; }
    v /= D_MAIN;
    sms[0] = m; sms[1] = rsqrtf(v + 1e-5f);
  }
  __syncthreads();
  sr[tid] = fmaxf((sx3[tid] - sms[0]) * sms[1] * ln2s[tid] + ln2b[tid], 0.f);
  __syncthreads();
  if (tid < N_CLASSES) {
    float a = Pb[tid];
    for (int k = 0; k < D_MAIN; ++k) a += sr[k] * PW[(size_t)k * N_CLASSES + tid];
    out[(size_t)b * N_CLASSES + tid] = a;
  }
}

// ---------------------------------------------------------------------------
// Host-side launcher
// ---------------------------------------------------------------------------
extern "C" void kernel_launch(void* const* d_in, const int* in_sizes, int n_in,
                              void* d_out, int out_size, void* d_ws, size_t ws_size,
                              hipStream_t stream)
{
  const float* x_num    = (const float*)d_in[0];
  const float* cand_num = (const float*)d_in[1];
  const int*   cand_y   = (const int*)d_in[2];
  // d_in[3] = context_size scalar (compile-time CTX)
  const float* W_lin = (const float*)d_in[4];
  const float* b_lin = (const float*)d_in[5];
  const float* W_K   = (const float*)d_in[6];
  const float* b_K   = (const float*)d_in[7];
  const float* Y_emb = (const float*)d_in[8];
  const float* T_W1  = (const float*)d_in[9];
  const float* T_b1  = (const float*)d_in[10];
  const float* T_W2  = (const float*)d_in[11];
  const float* bp_ln_s = (const float*)d_in[12];
  const float* bp_ln_b = (const float*)d_in[13];
  const float* bp_W1   = (const float*)d_in[14];
  const float* bp_b1   = (const float*)d_in[15];
  const float* bp_W2   = (const float*)d_in[16];
  const float* bp_b2   = (const float*)d_in[17];
  const float* P_ln_s  = (const float*)d_in[18];
  const float* P_ln_b  = (const float*)d_in[19];
  const float* P_W     = (const float*)d_in[20];
  const float* P_b     = (const float*)d_in[21];

  const int nb = in_sizes[0] / N_FEAT;   // 1024 queries
  const int nc = in_sizes[1] / N_FEAT;   // 100000 candidates

  // workspace carve (16B-aligned): x, k, ki_all, ||ki||^2, topi, x+V, partials
  float* ws  = (float*)d_ws;
  float* xq  = ws;
  float* kq  = xq + (size_t)nb * D_MAIN;
  float* ki  = kq + (size_t)nb * D_MAIN;
  float* cn  = ki + (size_t)nc * D_MAIN;
  int*  topi = (int*)(cn + nc);
  float* x2  = (float*)(topi + (size_t)nb * CTX);
  float* pdist = x2 + (size_t)nb * D_MAIN;
  int*   pidx  = (int*)(pdist + (size_t)nb * NPART * CTX);

  encode_kernel<<<nb / 16, 128, 0, stream>>>(x_num, W_lin, b_lin, W_K, b_K,
                                             xq, kq, nb);
  encode_kernel<<<(nc + 15) / 16, 128, 0, stream>>>(cand_num, W_lin, b_lin,
                                                    W_K, b_K, nullptr, ki, nc);
  row_sumsq_kernel<<<(nc + 255) / 256, 256, 0, stream>>>(ki, cn, nc);

  const size_t topk_lds = (size_t)(16 + 128) * LDA * 4 + (size_t)16 * LDH * 4 +
                          (size_t)16 * CTX * 4 * 2 + 64;
  topk_part_kernel<<<dim3(nb / 16, NPART), 128, topk_lds, stream>>>(
      kq, ki, cn, pdist, pidx, nc, NPART);
  topk_merge_kernel<<<nb, 64, 0, stream>>>(pdist, pidx, topi, NPART);

  const size_t agg_lds = (size_t)96 * LDA * 4 + (size_t)96 * LDH * 4 +
                         (size_t)D_MAIN * 4 + (size_t)4 * CTX * 4 + 64;
  aggregate_kernel<<<nb, 256, agg_lds, stream>>>(xq, kq, ki, topi, cand_y,
                                                 Y_emb, T_W1, T_b1, T_W2, x2);

  head_kernel<<<nb, 256, 0, stream>>>(x2, bp_ln_s, bp_ln_b, bp_W1, bp_b1,
                                      bp_W2, bp_b2, P_ln_s, P_ln_b, P_W, P_b,
                                      (float*)d_out);
}